// Block_ATC_21689584844855
// MI455X (gfx1250) — compile-verified
//
#include <hip/hip_runtime.h>
#include <hip/hip_bf16.h>
#include <math.h>

#define Bn 8
#define Tn 1024
#define Cn 192

typedef __attribute__((ext_vector_type(2))) float v2f;
typedef __attribute__((ext_vector_type(8))) float v8f;

// ---------------------------------------------------------------------------
// Kernel 1: row-normalize x -> xn.  One wave (32 lanes) per row of 192 floats.
// ---------------------------------------------------------------------------
__global__ __launch_bounds__(256) void normalize_rows(const float* __restrict__ x,
                                                      float* __restrict__ xn) {
    const int wave = threadIdx.x >> 5;
    const int lane = threadIdx.x & 31;
    const int row  = blockIdx.x * 8 + wave;          // Bn*Tn rows, grid sized exactly
    const float* src = x + (size_t)row * Cn;
    float vals[6];
    float s = 0.0f;
#pragma unroll
    for (int i = 0; i < 6; ++i) {
        vals[i] = src[lane + 32 * i];
        s += vals[i] * vals[i];
    }
#pragma unroll
    for (int off = 16; off > 0; off >>= 1) s += __shfl_xor(s, off, 32);
    const float inv = 1.0f / sqrtf(s);
    float* dst = xn + (size_t)row * Cn;
#pragma unroll
    for (int i = 0; i < 6; ++i) dst[lane + 32 * i] = vals[i] * inv;
}

// ---------------------------------------------------------------------------
// Kernel 2: dist = 1 - xn @ xn^T per batch, fp32 WMMA (V_WMMA_F32_16X16X4_F32).
// One wave computes one 16x16 tile; 48 WMMA ops over K=192.
// Writes both d_out (pristine dist) and Dw (mutable copy for clustering).
//
// f32 16x16x4 operand layout (ISA 7.12.2):
//   A (16x4): lane L holds A[L%16][2*(L/16) + {0,1}]  (2 VGPRs)
//   B (4x16): lane L holds B[2*(L/16) + {0,1}][L%16]  (2 VGPRs)
//   C/D     : VGPR v, lane L -> row v + 8*(L/16), col L%16
// Since B[k][n] = xn[tile_n + n][k], both A and B use the same addressing with
// different row bases.
// ---------------------------------------------------------------------------
__global__ __launch_bounds__(256) void gemm_dist(const float* __restrict__ xn,
                                                 float* __restrict__ dist,
                                                 float* __restrict__ Dw) {
    const int wave = threadIdx.x >> 5;
    const int lane = threadIdx.x & 31;
    const int tile = blockIdx.x * 8 + wave;          // Bn*64*64 tiles, exact grid
    const int b    = tile >> 12;                     // 4096 tiles per batch
    const int rem  = tile & 4095;
    const int tm   = (rem >> 6) << 4;
    const int tn   = (rem & 63) << 4;

    const float* xb = xn + (size_t)b * (Tn * Cn);
    const int lm   = lane & 15;
    const int koff = (lane >> 4) << 1;               // 0 or 2
    const float* pa = xb + (size_t)(tm + lm) * Cn + koff;
    const float* pb = xb + (size_t)(tn + lm) * Cn + koff;

    v8f acc = {};
#pragma unroll
    for (int k = 0; k < Cn; k += 4) {
        v2f a  = *(const v2f*)(pa + k);
        v2f bv = *(const v2f*)(pb + k);
        acc = __builtin_amdgcn_wmma_f32_16x16x4_f32(
            /*neg_a=*/false, a, /*neg_b=*/false, bv,
            /*c_mod=*/(short)0, acc, /*reuse_a=*/false, /*reuse_b=*/false);
    }

    const size_t base = (size_t)b * Tn * Tn;
    const int col   = tn + lm;
    const int rbase = tm + ((lane >> 4) << 3);
#pragma unroll
    for (int v = 0; v < 8; ++v) {
        const float dv = 1.0f - acc[v];
        const size_t idx = base + (size_t)(rbase + v) * Tn + col;
        dist[idx] = dv;
        Dw[idx]   = dv;
    }
}

// ---------------------------------------------------------------------------
// Kernel 3: average-linkage agglomerative clustering. One 1024-thread block
// (32 waves) per batch; thread t owns row t. Per-row NN caches in LDS avoid a
// full 1M-element argmin per merge.
// ---------------------------------------------------------------------------
__global__ __launch_bounds__(1024) void cluster_kernel(float* __restrict__ Dw_all,
                                                       const int* __restrict__ p_ncl,
                                                       float* __restrict__ labels_out) {
    __shared__ float rowmin[Tn];
    __shared__ int   rowarg[Tn];
    __shared__ float sizes[Tn];
    __shared__ int   active[Tn];
    __shared__ int   labels[Tn];
    __shared__ float red_val[Tn];
    __shared__ int   red_row[Tn];
    __shared__ int   s_i, s_j;

    const int t = threadIdx.x;
    const int b = blockIdx.x;
    float* Dw = Dw_all + (size_t)b * Tn * Tn;
    const float INF = 1.0e9f;

    int ncl = p_ncl[0];
    if (ncl > Tn) ncl = Tn;
    if (ncl < 1)  ncl = 1;
    const int n_merges = Tn - ncl;

    sizes[t]  = 1.0f;
    active[t] = 1;
    labels[t] = t;

    // initial per-row nearest neighbor (smallest index wins ties)
    {
        float best = INF;
        int   barg = -1;
        const float* row = Dw + (size_t)t * Tn;
        for (int s = 0; s < Tn; ++s) {
            const float w = row[s];
            if (s != t && w < best) { best = w; barg = s; }
        }
        rowmin[t] = best;
        rowarg[t] = barg;
    }
    __syncthreads();

    for (int m = 0; m < n_merges; ++m) {
        // ---- global argmin over active rows: lexicographic (val, row) ----
        red_val[t] = active[t] ? rowmin[t] : INF;
        red_row[t] = t;
        __syncthreads();
        for (int off = 512; off > 0; off >>= 1) {
            if (t < off) {
                const float v2 = red_val[t + off];
                const int   r2 = red_row[t + off];
                if (v2 < red_val[t] || (v2 == red_val[t] && r2 < red_row[t])) {
                    red_val[t] = v2;
                    red_row[t] = r2;
                }
            }
            __syncthreads();
        }
        if (t == 0) { s_i = red_row[0]; s_j = rowarg[red_row[0]]; }
        __syncthreads();
        const int i = s_i;
        const int j = s_j;

        const float ni = sizes[i], nj = sizes[j];
        const float inv = 1.0f / (ni + nj);
        // read old rows fully before any write (avoid intra-block races)
        const float di = Dw[(size_t)i * Tn + t];
        const float dj = Dw[(size_t)j * Tn + t];
        const float v  = (ni * di + nj * dj) * inv;
        __syncthreads();
        Dw[(size_t)i * Tn + t] = v;      // row i
        Dw[(size_t)t * Tn + i] = v;      // column i
        if (t == 0) sizes[i] = ni + nj;
        if (t == j) { active[j] = 0; rowmin[j] = INF; rowarg[j] = -1; }
        if (labels[t] == j) labels[t] = i;
        __syncthreads();

        // ---- maintain per-row NN caches ----
        if (active[t]) {
            if (t == i) {
                float best = INF;
                int   barg = -1;
                const float* row = Dw + (size_t)i * Tn;
                for (int s = 0; s < Tn; ++s) {
                    const float w = row[s];
                    if (s != i && active[s] && w < best) { best = w; barg = s; }
                }
                rowmin[i] = best;
                rowarg[i] = barg;
            } else {
                // v is this row's new distance to merged cluster i
                if (v < rowmin[t] || (v == rowmin[t] && i < rowarg[t])) {
                    rowmin[t] = v;
                    rowarg[t] = i;
                } else if (rowarg[t] == i || rowarg[t] == j) {
                    float best = INF;
                    int   barg = -1;
                    const float* row = Dw + (size_t)t * Tn;
                    for (int s = 0; s < Tn; ++s) {
                        const float w = row[s];
                        if (s != t && active[s] && w < best) { best = w; barg = s; }
                    }
                    rowmin[t] = best;
                    rowarg[t] = barg;
                }
            }
        }
        __syncthreads();
    }

    // ---- compact labels: rank of representative among sorted active reps ----
    const int lab = labels[t];
    int rank = 0;
    for (int r = 0; r < lab; ++r) rank += active[r];
    labels_out[(size_t)b * Tn + t] = (float)rank;
}

// ---------------------------------------------------------------------------
extern "C" void kernel_launch(void* const* d_in, const int* in_sizes, int n_in,
                              void* d_out, int out_size, void* d_ws, size_t ws_size,
                              hipStream_t stream) {
    (void)in_sizes; (void)n_in; (void)out_size; (void)ws_size;
    const float* x     = (const float*)d_in[0];
    const int*   p_ncl = (const int*)d_in[1];

    float* out        = (float*)d_out;
    float* dist       = out;                              // Bn*Tn*Tn floats
    float* labels_out = out + (size_t)Bn * Tn * Tn;       // Bn*Tn floats

    float* xn = (float*)d_ws;                             // Bn*Tn*Cn floats
    float* Dw = xn + (size_t)Bn * Tn * Cn;                // Bn*Tn*Tn floats (mutable copy)

    normalize_rows<<<(Bn * Tn) / 8, 256, 0, stream>>>(x, xn);
    gemm_dist<<<(Bn * 64 * 64) / 8, 256, 0, stream>>>(xn, dist, Dw);
    cluster_kernel<<<Bn, 1024, 0, stream>>>(Dw, p_ncl, labels_out);
}